// pReLURT_45406394253469
// MI455X (gfx1250) — compile-verified
//
#include <hip/hip_runtime.h>
#include <cstdint>
#include <cstddef>

// ---------------- problem dims (fixed by the reference) ----------------
#define MD 8
#define ND 32
#define BD 256
#define CD 512
#define NBROWS (ND * BD)            // 8192 (N*B) rows of length C per m
#define ROWS_PER_BLOCK 64
#define THREADS 256
#define ITERS (ROWS_PER_BLOCK / 2)  // each iter: 256 float4 = 2 rows

typedef float v4f __attribute__((ext_vector_type(4)));  // native vector: OK for NT builtins

// ---------------- gfx1250 async-to-LDS path (guarded) ----------------
#if defined(__HIP_DEVICE_COMPILE__) && defined(__gfx1250__) &&              \
    __has_builtin(__builtin_amdgcn_global_load_async_to_lds_b32) &&          \
    __has_builtin(__builtin_amdgcn_s_wait_asynccnt)
#define USE_ASYNC_LDS 1
#else
#define USE_ASYNC_LDS 0
#endif

#if USE_ASYNC_LDS
typedef __attribute__((address_space(1))) int* gptr_i32;
typedef __attribute__((address_space(3))) int* lptr_i32;

__device__ __forceinline__ void async_copy_b32(const void* gsrc, void* ldst) {
    // Global AS1 / LDS AS3 pointers. addrspacecast flat->LDS is a low-32-bit
    // truncate on AMDGPU, so the integer round-trip reproduces it exactly.
    __builtin_amdgcn_global_load_async_to_lds_b32(
        (gptr_i32)(uintptr_t)gsrc,
        (lptr_i32)(uint32_t)(uintptr_t)ldst,
        /*imm offset*/0, /*cpol*/0);
}
#endif

__device__ __forceinline__ float softplus_raw(float x) {
    // log(1 + exp(x)) stable form; 1+exp(-|x|) in (1,2] so fast log is accurate.
    return fmaxf(x, 0.0f) + __logf(1.0f + __expf(-fabsf(x)));
}

__global__ __launch_bounds__(THREADS) void
eta_softplus_kernel(const float* __restrict__ z,
                    const int*   __restrict__ mask,
                    const float* __restrict__ eta,
                    float*       __restrict__ out)
{
    __shared__ int   s_mask[CD];        // mask slice for this m (2 KB)
    __shared__ float s_eta[13 * 5 + 3]; // 13x5 table (260 B)

    const int tid  = threadIdx.x;
    const int nblk = NBROWS / ROWS_PER_BLOCK;               // 128
    const int m    = blockIdx.x / nblk;
    const int nb0  = (blockIdx.x % nblk) * ROWS_PER_BLOCK;

    const int* gmask = mask + m * CD;

#if USE_ASYNC_LDS
    async_copy_b32(gmask + tid,           &s_mask[tid]);
    async_copy_b32(gmask + THREADS + tid, &s_mask[THREADS + tid]);
    if (tid < 65) async_copy_b32(eta + tid, &s_eta[tid]);
    __builtin_amdgcn_s_wait_asynccnt(0);
    __syncthreads();
#else
    s_mask[tid]           = gmask[tid];
    s_mask[THREADS + tid] = gmask[THREADS + tid];
    if (tid < 65) s_eta[tid] = eta[tid];
    __syncthreads();
#endif

    // Thread owns 4 consecutive c's; stride of 256 float4 (= 2 rows) keeps
    // c fixed, so the 20 coefficients are loop-invariant registers.
    const int c  = (tid & 127) << 2;
    const int r0 = tid >> 7;            // 0 or 1

    float e0[4], e1b[4], e2[4], e3[4], bt[4];
#pragma unroll
    for (int j = 0; j < 4; ++j) {
        const int    k = s_mask[c + j];
        const float* p = &s_eta[k * 5];
        const float  b = p[4];
        e0[j]  = p[0];
        e1b[j] = p[1] / b;              // fold the 1/beta into e1
        e2[j]  = p[2];
        e3[j]  = p[3];
        bt[j]  = b;
    }

    size_t off = ((size_t)m * NBROWS + (size_t)(nb0 + r0)) * CD + c;
    const size_t stride = 2u * CD;      // 1024 floats = 4 KB

#pragma unroll 4
    for (int it = 0; it < ITERS; ++it) {
        __builtin_prefetch(z + off + stride * 8, 0, 0);   // global_prefetch_b8
        // Streamed once: non-temporal (TH=NT) keeps the 256MB stream from
        // thrashing WGP$/L2 (§7.3 cache controls).
        const v4f v = __builtin_nontemporal_load((const v4f*)(z + off));
        v4f r;
        {
            const float zc = v.x - e2[0];
            r.x = fmaf(e0[0], zc, fmaf(e1b[0], softplus_raw(bt[0] * zc), e3[0]));
        }
        {
            const float zc = v.y - e2[1];
            r.y = fmaf(e0[1], zc, fmaf(e1b[1], softplus_raw(bt[1] * zc), e3[1]));
        }
        {
            const float zc = v.z - e2[2];
            r.z = fmaf(e0[2], zc, fmaf(e1b[2], softplus_raw(bt[2] * zc), e3[2]));
        }
        {
            const float zc = v.w - e2[3];
            r.w = fmaf(e0[3], zc, fmaf(e1b[3], softplus_raw(bt[3] * zc), e3[3]));
        }
        __builtin_nontemporal_store(r, (v4f*)(out + off));
        off += stride;
    }
}

extern "C" void kernel_launch(void* const* d_in, const int* in_sizes, int n_in,
                              void* d_out, int out_size, void* d_ws, size_t ws_size,
                              hipStream_t stream) {
    const float* z    = (const float*)d_in[0];
    const int*   mask = (const int*)d_in[1];
    const float* eta  = (const float*)d_in[2];
    float*       out  = (float*)d_out;

    const int grid = MD * (NBROWS / ROWS_PER_BLOCK);   // 8 * 128 = 1024 blocks
    eta_softplus_kernel<<<grid, THREADS, 0, stream>>>(z, mask, eta, out);

    (void)in_sizes; (void)n_in; (void)out_size; (void)d_ws; (void)ws_size;
}